// LIFcomplexLayer_42030549959284
// MI455X (gfx1250) — compile-verified
//
#include <hip/hip_runtime.h>

// ---------------------------------------------------------------------------
// LIF complex layer: GEMM (f32 WMMA, async-LDS double-buffered) -> BatchNorm
// (train stats) -> complex leaky-integrate scan with heaviside spikes.
// Shapes: x[32,2048,512], W[512,512], out[32,2048,512]
// ---------------------------------------------------------------------------

typedef float v2f __attribute__((ext_vector_type(2)));
typedef float v8f __attribute__((ext_vector_type(8)));

#define B_      32
#define T_      2048
#define I_      512
#define H_      512
#define M_TOTAL (B_ * T_)           // 65536 rows
#define RED_BLOCKS 256              // BN partial-reduction blocks
#define ROWS_PER_RED (M_TOTAL / RED_BLOCKS)
#define BN_EPS  1e-5f

#define M_BLK   128
#define N_BLK   64
#define KC      32                  // K-chunk staged in LDS
#define NCHUNK  (I_ / KC)           // 16
#define PADR    36                  // padded LDS row pitch (floats): bank-conflict-free

// Issue one async Global->LDS B128 copy per 16B element this thread owns.
// rows * KC floats staged; 8 x 16B chunks per row.
__device__ __forceinline__
void tile_async_load(int tid, const float* __restrict__ gbase, unsigned lds_base,
                     int rows)
{
    const int total = rows * 8;                 // 16B elements
    for (int e = tid; e < total; e += 256) {
        const int row = e >> 3;
        const int ch  = e & 7;
        const float* g = gbase + (size_t)row * I_ + ch * 4;
        const unsigned l = lds_base + (unsigned)(row * PADR + ch * 4) * 4u;
        asm volatile("global_load_async_to_lds_b128 %0, %1, off"
                     :: "v"(l), "v"(g) : "memory");
    }
}

__device__ __forceinline__ void wait_async_all()
{
    asm volatile("s_wait_asynccnt 0x0" ::: "memory");
}

// ---------------------------------------------------------------------------
// Kernel 1: Wx = x @ W^T  via V_WMMA_F32_16X16X4_F32, LDS double buffering.
//   block = 256 threads (8 waves), wave tile = 32x32 (2x2 of 16x16),
//   block tile = 128 (M) x 64 (N), grid = (512, 8)
// A-frag (16x4 f32): lane<16 -> M=lane, K={k,k+1}; lane>=16 -> K={k+2,k+3}
// B-frag (4x16 f32): lane<16 -> N=lane, K={k,k+1}; lane>=16 -> K={k+2,k+3}
// C/D: VGPR r, lane l -> row = r + 8*(l>>4), col = l&15
// ---------------------------------------------------------------------------
__global__ __launch_bounds__(256)
void gemm_wmma_f32(const float* __restrict__ x, const float* __restrict__ W,
                   float* __restrict__ Wx)
{
    __shared__ float Xs[2][M_BLK * PADR];   // 2 x 18432 B
    __shared__ float Ws[2][N_BLK * PADR];   // 2 x  9216 B

    const int tid   = threadIdx.x;
    const int lane  = tid & 31;
    const int wave  = tid >> 5;
    const int wm    = wave & 3;                    // wave row (0..3)
    const int wn    = wave >> 2;                   // wave col (0..1)
    const int m_base = blockIdx.x * M_BLK;
    const int n_base = blockIdx.y * N_BLK;
    const int m_off  = wm * 32;
    const int n_off  = wn * 32;
    const int lmod   = lane & 15;
    const int koff   = (lane >> 4) * 2;

    const unsigned xs0 = (unsigned)(unsigned long long)(const void*)&Xs[0][0];
    const unsigned xs1 = (unsigned)(unsigned long long)(const void*)&Xs[1][0];
    const unsigned ws0 = (unsigned)(unsigned long long)(const void*)&Ws[0][0];
    const unsigned ws1 = (unsigned)(unsigned long long)(const void*)&Ws[1][0];

    const float* xg = x + (size_t)m_base * I_;
    const float* wg = W + (size_t)n_base * I_;

    v8f acc00 = {}, acc01 = {}, acc10 = {}, acc11 = {};

    // preload chunk 0 into buffer 0
    tile_async_load(tid, xg, xs0, M_BLK);
    tile_async_load(tid, wg, ws0, N_BLK);
    wait_async_all();
    __syncthreads();

    for (int c = 0; c < NCHUNK; ++c) {
        const int buf = c & 1;
        if (c + 1 < NCHUNK) {   // prefetch next chunk into the other buffer
            tile_async_load(tid, xg + (c + 1) * KC, buf ? xs0 : xs1, M_BLK);
            tile_async_load(tid, wg + (c + 1) * KC, buf ? ws0 : ws1, N_BLK);
        }

        const float* __restrict__ xsb = &Xs[buf][0];
        const float* __restrict__ wsb = &Ws[buf][0];
#pragma unroll
        for (int k = 0; k < KC; k += 4) {
            v2f a0 = *(const v2f*)(xsb + (m_off      + lmod) * PADR + k + koff);
            v2f a1 = *(const v2f*)(xsb + (m_off + 16 + lmod) * PADR + k + koff);
            v2f b0 = *(const v2f*)(wsb + (n_off      + lmod) * PADR + k + koff);
            v2f b1 = *(const v2f*)(wsb + (n_off + 16 + lmod) * PADR + k + koff);
            acc00 = __builtin_amdgcn_wmma_f32_16x16x4_f32(false, a0, false, b0,
                                                          (short)0, acc00, false, false);
            acc01 = __builtin_amdgcn_wmma_f32_16x16x4_f32(false, a0, false, b1,
                                                          (short)0, acc01, false, false);
            acc10 = __builtin_amdgcn_wmma_f32_16x16x4_f32(false, a1, false, b0,
                                                          (short)0, acc10, false, false);
            acc11 = __builtin_amdgcn_wmma_f32_16x16x4_f32(false, a1, false, b1,
                                                          (short)0, acc11, false, false);
        }

        if (c + 1 < NCHUNK)
            wait_async_all();      // my prefetch writes have landed in LDS
        __syncthreads();           // everyone done reading buf & writing buf^1
    }

    const int row_off = (lane >> 4) * 8;
#pragma unroll
    for (int r = 0; r < 8; ++r) {
        const size_t m0 = (size_t)(m_base + m_off + row_off + r);
        const size_t m1 = m0 + 16;
        Wx[m0 * H_ + n_base + n_off + lmod]      = acc00[r];
        Wx[m0 * H_ + n_base + n_off + 16 + lmod] = acc01[r];
        Wx[m1 * H_ + n_base + n_off + lmod]      = acc10[r];
        Wx[m1 * H_ + n_base + n_off + 16 + lmod] = acc11[r];
    }
}

// ---------------------------------------------------------------------------
// Kernel 2: per-column partial sums / sums-of-squares (deterministic: no atomics)
// ---------------------------------------------------------------------------
__global__ __launch_bounds__(512)
void bn_partial(const float* __restrict__ Wx,
                float* __restrict__ psum, float* __restrict__ psq)
{
    const int h   = threadIdx.x;           // 0..511
    const int blk = blockIdx.x;            // 0..255
    const size_t row0 = (size_t)blk * ROWS_PER_RED;
    float s = 0.f, q = 0.f;
    for (int r = 0; r < ROWS_PER_RED; ++r) {
        float v = Wx[(row0 + r) * H_ + h];
        s += v;
        q += v * v;
    }
    psum[blk * H_ + h] = s;
    psq [blk * H_ + h] = q;
}

// ---------------------------------------------------------------------------
// Kernel 3: finalize BN (scale/shift) + precompute complex decay alpha per h
// ---------------------------------------------------------------------------
__global__ __launch_bounds__(512)
void bn_finalize(const float* __restrict__ psum, const float* __restrict__ psq,
                 const float* __restrict__ gamma, const float* __restrict__ beta,
                 const float* __restrict__ lla,   const float* __restrict__ ldt,
                 const float* __restrict__ aimg,
                 float* __restrict__ scale, float* __restrict__ shift,
                 float* __restrict__ ar,    float* __restrict__ ai)
{
    const int h = threadIdx.x;             // 0..511
    float s = 0.f, q = 0.f;
    for (int i = 0; i < RED_BLOCKS; ++i) {
        s += psum[i * H_ + h];
        q += psq [i * H_ + h];
    }
    const float inv_n = 1.0f / (float)M_TOTAL;
    const float mean  = s * inv_n;
    const float var   = q * inv_n - mean * mean;
    const float sc    = gamma[h] * rsqrtf(var + BN_EPS);
    scale[h] = sc;
    shift[h] = beta[h] - mean * sc;

    // alpha = exp((-exp(lla) + i*aimg) * exp(ldt))
    const float dt  = expf(ldt[h]);
    const float mag = expf(-expf(lla[h]) * dt);
    const float ang = aimg[h] * dt;
    ar[h] = mag * cosf(ang);
    ai[h] = mag * sinf(ang);
}

// ---------------------------------------------------------------------------
// Kernel 4: sequential complex LIF scan over T, thread per (b,h)
// ---------------------------------------------------------------------------
__global__ __launch_bounds__(256)
void lif_scan(const float* __restrict__ Wx,
              const float* __restrict__ scale, const float* __restrict__ shift,
              const float* __restrict__ arr,   const float* __restrict__ aii,
              const float* __restrict__ bcoef,
              const float* __restrict__ u0r, const float* __restrict__ u0i,
              const float* __restrict__ s0,
              float* __restrict__ out)
{
    const int b = blockIdx.x >> 1;                          // 0..31
    const int h = ((blockIdx.x & 1) << 8) + threadIdx.x;    // 0..511

    const float sc = scale[h], sh = shift[h];
    const float ar = arr[h],  ai = aii[h], bh = bcoef[h];

    float ur = u0r[b * H_ + h];
    float ui = u0i[b * H_ + h];
    float s  = s0 [b * H_ + h];

    const float* wxp = Wx  + (size_t)b * T_ * H_ + h;
    float*       op  = out + (size_t)b * T_ * H_ + h;

    for (int t = 0; t < T_; ++t) {
        if (t + 16 < T_)   // emits global_prefetch_b8; hides stride-2KB latency
            __builtin_prefetch(wxp + (size_t)(t + 16) * H_, 0, 0);

        const float wx = wxp[(size_t)t * H_] * sc + sh;
        const float tr = ur - s;        // (u - s).real
        const float ti = ui;            // (u - s).imag
        ur = ar * tr - ai * ti + bh * wx;
        ui = ar * ti + ai * tr;
        s  = (2.0f * ur - 1.0f) > 0.0f ? 1.0f : 0.0f;
        op[(size_t)t * H_] = s;
    }
}

// ---------------------------------------------------------------------------
// Launch
// ---------------------------------------------------------------------------
extern "C" void kernel_launch(void* const* d_in, const int* in_sizes, int n_in,
                              void* d_out, int out_size, void* d_ws, size_t ws_size,
                              hipStream_t stream)
{
    const float* x     = (const float*)d_in[0];   // [B,T,I]
    const float* W     = (const float*)d_in[1];   // [H,I]
    const float* lla   = (const float*)d_in[2];   // [H]
    const float* ldt   = (const float*)d_in[3];   // [H]
    const float* aimg  = (const float*)d_in[4];   // [H]
    const float* bcoef = (const float*)d_in[5];   // [H]
    const float* gamma = (const float*)d_in[6];   // [H]
    const float* beta  = (const float*)d_in[7];   // [H]
    const float* u0r   = (const float*)d_in[8];   // [B,H]
    const float* u0i   = (const float*)d_in[9];   // [B,H]
    const float* s0    = (const float*)d_in[10];  // [B,H]
    float* out = (float*)d_out;

    float* ws    = (float*)d_ws;
    float* Wx    = ws;                                   // 65536*512 floats
    float* psum  = Wx   + (size_t)M_TOTAL * H_;          // 256*512
    float* psq   = psum + (size_t)RED_BLOCKS * H_;       // 256*512
    float* scale = psq  + (size_t)RED_BLOCKS * H_;       // 512
    float* shift = scale + H_;                           // 512
    float* ar    = shift + H_;                           // 512
    float* ai    = ar    + H_;                           // 512

    dim3 gemm_grid(M_TOTAL / M_BLK, H_ / N_BLK);         // (512, 8)
    gemm_wmma_f32<<<gemm_grid, 256, 0, stream>>>(x, W, Wx);

    bn_partial<<<RED_BLOCKS, 512, 0, stream>>>(Wx, psum, psq);

    bn_finalize<<<1, 512, 0, stream>>>(psum, psq, gamma, beta,
                                       lla, ldt, aimg, scale, shift, ar, ai);

    lif_scan<<<B_ * 2, 256, 0, stream>>>(Wx, scale, shift, ar, ai, bcoef,
                                         u0r, u0i, s0, out);
}